// OurConvNetcell_26706106647093
// MI455X (gfx1250) — compile-verified
//
#include <hip/hip_runtime.h>

#define FD 128
#define NEG_SLOPE 0.1f
#define BN_EPS 1e-5f

typedef float v2f __attribute__((ext_vector_type(2)));
typedef float v8f __attribute__((ext_vector_type(8)));

__device__ __forceinline__ float leakyf(float v) {
    return v > 0.0f ? v : NEG_SLOPE * v;
}

// ---------------------------------------------------------------- zero fill
__global__ void zero_f4(float4* __restrict__ p, long n4) {
    long i = (long)blockIdx.x * blockDim.x + threadIdx.x;
    long stride = (long)gridDim.x * blockDim.x;
    float4 z = make_float4(0.f, 0.f, 0.f, 0.f);
    for (; i < n4; i += stride) p[i] = z;
}

// ---------------------------------------------------------------- in-degree
__global__ void deg_kernel(const int* __restrict__ ee, float* __restrict__ degf, int E) {
    int i = blockIdx.x * blockDim.x + threadIdx.x;
    if (i < E) atomicAdd(&degf[ee[i]], 1.0f);
}

// ---------------------------------------------------------------- fused GEMM: O{0,1,2} = H @ W{0,1,2}^T
// One wave = one 16x16 output tile; f32 WMMA 16x16x4, K = 128.
// A fragment: lanes 0-15 rows m0..m0+15 (K=k0,k0+1); lanes 16-31 same rows (K=k0+2,k0+3).
// B fragment: lane = output column; B[k][n] = W[n][k]  (out = H * W^T).
__global__ void gemm3_wmma(const float* __restrict__ H,
                           const float* __restrict__ W0,
                           const float* __restrict__ W1,
                           const float* __restrict__ W2,
                           float* __restrict__ O0, float* __restrict__ O1,
                           float* __restrict__ O2, int Nrows) {
    const int wave = threadIdx.x >> 5;          // 0..7 -> column tile
    const int lane = threadIdx.x & 31;
    const int half = lane >> 4;                 // K-half select
    const int lrow = lane & 15;
    const int m0 = blockIdx.x * 16;
    const int n0 = wave * 16;
    if (m0 >= Nrows) return;                    // wave-uniform

    const float* ha = H  + (size_t)(m0 + lrow) * FD;
    const float* w0 = W0 + (size_t)(n0 + lrow) * FD;
    const float* w1 = W1 + (size_t)(n0 + lrow) * FD;
    const float* w2 = W2 + (size_t)(n0 + lrow) * FD;

    v8f acc0 = {}; v8f acc1 = {}; v8f acc2 = {};
#pragma unroll 8
    for (int k0 = 0; k0 < FD; k0 += 4) {
        const int kb = k0 + (half << 1);
        v2f a  = *(const v2f*)(ha + kb);
        v2f b0 = *(const v2f*)(w0 + kb);
        v2f b1 = *(const v2f*)(w1 + kb);
        v2f b2 = *(const v2f*)(w2 + kb);
        acc0 = __builtin_amdgcn_wmma_f32_16x16x4_f32(false, a, false, b0, (short)0, acc0, false, false);
        acc1 = __builtin_amdgcn_wmma_f32_16x16x4_f32(false, a, false, b1, (short)0, acc1, false, false);
        acc2 = __builtin_amdgcn_wmma_f32_16x16x4_f32(false, a, false, b2, (short)0, acc2, false, false);
    }
    // D layout: VGPR r -> row m0 + r + 8*half, col n0 + lrow
    size_t obase = (size_t)(m0 + half * 8) * FD + n0 + lrow;
#pragma unroll
    for (int r = 0; r < 8; ++r) {
        O0[obase + (size_t)r * FD] = acc0[r];
        O1[obase + (size_t)r * FD] = acc1[r];
        O2[obase + (size_t)r * FD] = acc2[r];
    }
}

// ---------------------------------------------------------------- final GEMM with residual + leaky
// out = leaky( X @ R^T + H2 )
__global__ void gemm_res_wmma(const float* __restrict__ X,
                              const float* __restrict__ Rw,
                              const float* __restrict__ H2,
                              float* __restrict__ out, int Nrows) {
    const int wave = threadIdx.x >> 5;
    const int lane = threadIdx.x & 31;
    const int half = lane >> 4;
    const int lrow = lane & 15;
    const int m0 = blockIdx.x * 16;
    const int n0 = wave * 16;
    if (m0 >= Nrows) return;

    const float* ha = X  + (size_t)(m0 + lrow) * FD;
    const float* wr = Rw + (size_t)(n0 + lrow) * FD;

    v8f acc = {};
#pragma unroll 8
    for (int k0 = 0; k0 < FD; k0 += 4) {
        const int kb = k0 + (half << 1);
        v2f a = *(const v2f*)(ha + kb);
        v2f b = *(const v2f*)(wr + kb);
        acc = __builtin_amdgcn_wmma_f32_16x16x4_f32(false, a, false, b, (short)0, acc, false, false);
    }
    size_t obase = (size_t)(m0 + half * 8) * FD + n0 + lrow;
#pragma unroll
    for (int r = 0; r < 8; ++r) {
        float v = acc[r] + H2[obase + (size_t)r * FD];
        out[obase + (size_t)r * FD] = leakyf(v);
    }
}

// ---------------------------------------------------------------- edge gather/gate/scatter
// One wave per edge, each lane handles 4 consecutive features (32*4 = 128).
__global__ void edge_gate_scatter(const float* __restrict__ Uix,
                                  const float* __restrict__ Vix,
                                  const float* __restrict__ Vjx,
                                  const float* __restrict__ bv,
                                  const int* __restrict__ es,
                                  const int* __restrict__ ee,
                                  float* __restrict__ agg, int E) {
    int gw = (blockIdx.x * blockDim.x + threadIdx.x) >> 5;
    int lane = threadIdx.x & 31;
    if (gw >= E) return;
    int s = es[gw];
    int t = ee[gw];
    float4 u  = ((const float4*)(Uix + (size_t)s * FD))[lane];
    float4 vi = ((const float4*)(Vix + (size_t)t * FD))[lane];
    float4 vj = ((const float4*)(Vjx + (size_t)s * FD))[lane];
    float4 bb = ((const float4*)bv)[lane];
    float* dst = agg + (size_t)t * FD + lane * 4;
    float g;
    g = 1.f / (1.f + __expf(-(vi.x + vj.x + bb.x))); atomicAdd(dst + 0, g * u.x);
    g = 1.f / (1.f + __expf(-(vi.y + vj.y + bb.y))); atomicAdd(dst + 1, g * u.y);
    g = 1.f / (1.f + __expf(-(vi.z + vj.z + bb.z))); atomicAdd(dst + 2, g * u.z);
    g = 1.f / (1.f + __expf(-(vi.w + vj.w + bb.w))); atomicAdd(dst + 3, g * u.w);
}

// ---------------------------------------------------------------- t = (agg + bu)/norm in-place + column sum/sumsq
__global__ void finish_stats(float* __restrict__ tmat, const float* __restrict__ bu,
                             const float* __restrict__ degf,
                             float* __restrict__ colsum, float* __restrict__ colsumsq,
                             int Nrows) {
    int d = threadIdx.x;   // 0..127, one column per thread
    float bud = bu[d];
    float s = 0.f, s2 = 0.f;
    for (int n = blockIdx.x; n < Nrows; n += gridDim.x) {
        float inv = 1.f / fmaxf(degf[n], 1.f);
        float t = (tmat[(size_t)n * FD + d] + bud) * inv;
        tmat[(size_t)n * FD + d] = t;
        s += t;
        s2 += t * t;
    }
    atomicAdd(&colsum[d], s);
    atomicAdd(&colsumsq[d], s2);
}

// ---------------------------------------------------------------- fold BN stats into scale/shift
__global__ void bn_stats(const float* __restrict__ colsum, const float* __restrict__ colsumsq,
                         const float* __restrict__ gamma, const float* __restrict__ beta,
                         float* __restrict__ scale, float* __restrict__ shift, float invN) {
    int d = threadIdx.x;
    float mean = colsum[d] * invN;
    float var  = colsumsq[d] * invN - mean * mean;   // biased variance
    float sc = gamma[d] * rsqrtf(var + BN_EPS);
    scale[d] = sc;
    shift[d] = beta[d] - mean * sc;
}

// ---------------------------------------------------------------- apply BN (+ optional leaky) in-place
__global__ void bn_apply(float* __restrict__ tmat, const float* __restrict__ scale,
                         const float* __restrict__ shift, int do_leaky, long n4) {
    long i = (long)blockIdx.x * blockDim.x + threadIdx.x;
    long stride = (long)gridDim.x * blockDim.x;
    for (; i < n4; i += stride) {
        int c = (int)(i & 31);             // 32 float4 per row
        float4 t  = ((float4*)tmat)[i];
        float4 sc = ((const float4*)scale)[c];
        float4 sh = ((const float4*)shift)[c];
        t.x = t.x * sc.x + sh.x;
        t.y = t.y * sc.y + sh.y;
        t.z = t.z * sc.z + sh.z;
        t.w = t.w * sc.w + sh.w;
        if (do_leaky) {
            t.x = leakyf(t.x); t.y = leakyf(t.y);
            t.z = leakyf(t.z); t.w = leakyf(t.w);
        }
        ((float4*)tmat)[i] = t;
    }
}

// ---------------------------------------------------------------- host launcher
extern "C" void kernel_launch(void* const* d_in, const int* in_sizes, int n_in,
                              void* d_out, int out_size, void* d_ws, size_t ws_size,
                              hipStream_t stream) {
    const float* x    = (const float*)d_in[0];
    const float* Ui1  = (const float*)d_in[1];
    const float* Vi1  = (const float*)d_in[2];
    const float* Vj1  = (const float*)d_in[3];
    const float* bu1  = (const float*)d_in[4];
    const float* bv1  = (const float*)d_in[5];
    const float* Ui2  = (const float*)d_in[6];
    const float* Vi2  = (const float*)d_in[7];
    const float* Vj2  = (const float*)d_in[8];
    const float* bu2  = (const float*)d_in[9];
    const float* bv2  = (const float*)d_in[10];
    const float* g1   = (const float*)d_in[11];
    const float* b1   = (const float*)d_in[12];
    const float* g2   = (const float*)d_in[13];
    const float* b2   = (const float*)d_in[14];
    const float* Rw   = (const float*)d_in[15];
    const int*   es   = (const int*)d_in[16];
    const int*   ee   = (const int*)d_in[17];

    const int N = in_sizes[0] / FD;
    const int E = in_sizes[16];
    const size_t ND = (size_t)N * FD;

    // workspace layout (floats): 4 N*D buffers + degree + BN stats
    float* A0       = (float*)d_ws;
    float* A1       = A0 + ND;
    float* A2       = A1 + ND;
    float* A3       = A2 + ND;     // agg / t / h buffer
    float* degf     = A3 + ND;
    float* colsum   = degf + N;
    float* colsumsq = colsum + FD;
    float* scale    = colsumsq + FD;
    float* shift    = scale + FD;

    const int rowTiles   = (N + 15) / 16;          // 3125 for N=50000
    const dim3 gemmGrid(rowTiles), gemmBlk(256);   // 8 waves = 8 col tiles
    const int edgeBlocks = (E + 7) / 8;            // 8 edges (waves) per 256-thr block
    const long nd4 = (long)(ND / 4);
    const float invN = 1.0f / (float)N;

    // ---- degree (shared by both layers)
    zero_f4<<<64, 256, 0, stream>>>((float4*)degf, (long)(N / 4));
    deg_kernel<<<(E + 255) / 256, 256, 0, stream>>>(ee, degf, E);

    // ================= layer 1 =================
    gemm3_wmma<<<gemmGrid, gemmBlk, 0, stream>>>(x, Ui1, Vi1, Vj1, A0, A1, A2, N);
    zero_f4<<<2048, 256, 0, stream>>>((float4*)A3, nd4);
    edge_gate_scatter<<<edgeBlocks, 256, 0, stream>>>(A0, A1, A2, bv1, es, ee, A3, E);
    zero_f4<<<1, 64, 0, stream>>>((float4*)colsum, 64);           // colsum+colsumsq (256 f)
    finish_stats<<<512, FD, 0, stream>>>(A3, bu1, degf, colsum, colsumsq, N);
    bn_stats<<<1, FD, 0, stream>>>(colsum, colsumsq, g1, b1, scale, shift, invN);
    bn_apply<<<2048, 256, 0, stream>>>(A3, scale, shift, /*leaky=*/1, nd4);   // h1 in A3

    // ================= layer 2 =================
    gemm3_wmma<<<gemmGrid, gemmBlk, 0, stream>>>(A3, Ui2, Vi2, Vj2, A0, A1, A2, N);
    zero_f4<<<2048, 256, 0, stream>>>((float4*)A3, nd4);          // h1 consumed; reuse as agg
    edge_gate_scatter<<<edgeBlocks, 256, 0, stream>>>(A0, A1, A2, bv2, es, ee, A3, E);
    zero_f4<<<1, 64, 0, stream>>>((float4*)colsum, 64);
    finish_stats<<<512, FD, 0, stream>>>(A3, bu2, degf, colsum, colsumsq, N);
    bn_stats<<<1, FD, 0, stream>>>(colsum, colsumsq, g2, b2, scale, shift, invN);
    bn_apply<<<2048, 256, 0, stream>>>(A3, scale, shift, /*leaky=*/0, nd4);   // h2 in A3

    // ================= residual head: out = leaky(x @ R^T + h2) =================
    gemm_res_wmma<<<gemmGrid, gemmBlk, 0, stream>>>(x, Rw, A3, (float*)d_out, N);
}